// VectorQuantizer_13950053777814
// MI455X (gfx1250) — compile-verified
//
#include <hip/hip_runtime.h>
#include <stdint.h>

typedef float v2f __attribute__((ext_vector_type(2)));
typedef float v8f __attribute__((ext_vector_type(8)));

#define Bsz 16
#define Csz 64
#define Hsz 64
#define Wsz 64
#define Ksz 1024
#define HW   (Hsz*Wsz)        // 4096
#define CHW  (Csz*HW)         // 262144
#define Nflat (Bsz*HW)        // 65536

// ---------------------------------------------------------------------------
// Kernel 1: ||e_k||^2 for each codebook row. One wave32 per row.
// ---------------------------------------------------------------------------
__global__ void __launch_bounds__(256) vq_norms(const float* __restrict__ emb,
                                                float* __restrict__ norms) {
    int t    = blockIdx.x * 256 + threadIdx.x;
    int wave = t >> 5;          // codebook row, 0..1023
    int lane = t & 31;
    const float* row = emb + wave * Csz;
    float a = row[lane];
    float b = row[lane + 32];
    float s = a * a + b * b;
    #pragma unroll
    for (int off = 16; off > 0; off >>= 1)
        s += __shfl_xor(s, off, 32);
    if (lane == 0) norms[wave] = s;
}

// ---------------------------------------------------------------------------
// Kernel 2: fused GEMM + argmin via V_WMMA_F32_16X16X4_F32.
// Whole 256KB codebook staged in LDS (CDNA5 WGP has 320KB) using the
// CDNA5 async memory->LDS path (GLOBAL_LOAD_ASYNC_TO_LDS_B128, ASYNCcnt):
// no VGPR round-trip for the staging copy.
// Each wave handles one 16-row tile of flattened latents.
// ---------------------------------------------------------------------------
__global__ void __launch_bounds__(256) vq_argmin(const float* __restrict__ in,
                                                 const float* __restrict__ emb,
                                                 const float* __restrict__ norms,
                                                 int* __restrict__ idxout) {
    extern __shared__ float lds[];
    float* e_lds = lds;                  // 1024*64 floats (256 KB)
    float* n_lds = lds + Ksz * Csz;      // 1024 floats (4 KB)
    const int t = threadIdx.x;

    // ---- Async staging: codebook (16384 x b128) + norms (1024 x b32) ----
    // Generic shared-space address low 32 bits == wave-relative LDS offset
    // (aperture spec).
    {
        const unsigned elds_base = (unsigned)(uintptr_t)e_lds;
        const unsigned nlds_base = (unsigned)(uintptr_t)n_lds;
        const char* esrc = (const char*)emb;
        const char* nsrc = (const char*)norms;
        #pragma unroll 4
        for (int i = 0; i < (Ksz * Csz * 4) / (256 * 16); ++i) {  // 64 iters
            unsigned loff = elds_base + (unsigned)(t * 16 + i * 4096);
            uint64_t gaddr = (uint64_t)(esrc + t * 16 + i * 4096);
            asm volatile("global_load_async_to_lds_b128 %0, %1, off"
                         :: "v"(loff), "v"(gaddr) : "memory");
        }
        #pragma unroll
        for (int i = 0; i < Ksz / 256; ++i) {                     // 4 iters
            unsigned loff = nlds_base + (unsigned)(t * 4 + i * 1024);
            uint64_t gaddr = (uint64_t)(nsrc + t * 4 + i * 1024);
            asm volatile("global_load_async_to_lds_b32 %0, %1, off"
                         :: "v"(loff), "v"(gaddr) : "memory");
        }
        asm volatile("s_wait_asynccnt 0" ::: "memory");
    }
    __syncthreads();

    const int wave = t >> 5;
    const int lane = t & 31;
    const int hi   = lane >> 4;     // half-wave select (ISA A/B layout)
    const int m    = lane & 15;     // row within tile / column within B group
    const int tile  = blockIdx.x * 8 + wave;
    const int nbase = tile * 16;                // 16 | W so tile shares (b,h)
    const int b   = nbase >> 12;
    const int rem = nbase & 4095;
    const int h   = rem >> 6;
    const int w0  = rem & 63;
    const float* Abase = in + b * CHW + h * Wsz + w0 + m;

    // A-tile (16x64 fp32) resident in 32 VGPRs, ISA 16x4 fp32 A layout:
    // lanes 0-15 hold K = 4k0+{0,1}; lanes 16-31 hold K = 4k0+{2,3}.
    v2f a_reg[16];
    #pragma unroll
    for (int k0 = 0; k0 < 16; ++k0) {
        int c0 = 4 * k0 + 2 * hi;
        a_reg[k0][0] = Abase[(c0 + 0) * HW];
        a_reg[k0][1] = Abase[(c0 + 1) * HW];
    }

    float best[8];
    int   bidx[8];
    #pragma unroll
    for (int v = 0; v < 8; ++v) { best[v] = 3.4e38f; bidx[v] = 0; }

    // Sweep 64 groups of 16 embeddings; D = A x B accumulated over 16 K-chunks.
    for (int n0 = 0; n0 < Ksz / 16; ++n0) {
        v8f acc = {};
        const float* brow = e_lds + (n0 * 16 + m) * Csz + 2 * hi;
        #pragma unroll
        for (int k0 = 0; k0 < 16; ++k0) {
            v2f bv = *(const v2f*)(brow + 4 * k0);   // ds_load_2addr_b64 pairs
            acc = __builtin_amdgcn_wmma_f32_16x16x4_f32(
                false, a_reg[k0], false, bv, (short)0, acc, false, false);
        }
        // score = ||e||^2 - 2 f.e   (||f||^2 constant per row -> dropped)
        const float nrm  = n_lds[n0 * 16 + m];
        const int   kidx = n0 * 16 + m;
        #pragma unroll
        for (int v = 0; v < 8; ++v) {
            float s = nrm - 2.0f * acc[v];
            if (s < best[v]) { best[v] = s; bidx[v] = kidx; }  // strict <: first idx wins
        }
    }

    // Reduce (score,idx) across the 16 lanes of each half-wave
    // (C/D layout: VGPR v holds row M=v for lanes 0-15, M=v+8 for 16-31).
    #pragma unroll
    for (int off = 8; off >= 1; off >>= 1) {
        #pragma unroll
        for (int v = 0; v < 8; ++v) {
            float os = __shfl_xor(best[v], off, 32);
            int   oi = __shfl_xor(bidx[v], off, 32);
            if (os < best[v] || (os == best[v] && oi < bidx[v])) {
                best[v] = os; bidx[v] = oi;
            }
        }
    }
    if (m == 0) {   // lanes 0 and 16 publish rows M=0..7 and M=8..15
        #pragma unroll
        for (int v = 0; v < 8; ++v)
            idxout[nbase + 8 * hi + v] = bidx[v];
    }
}

// ---------------------------------------------------------------------------
// Kernel 3: per-(b,h) 64x64 LDS-tiled transpose + codebook gather.
// Produces quantized [B,C,H,W], latents [B,H,W,C], quantized_latents [B,H,W,C].
// ---------------------------------------------------------------------------
__global__ void __launch_bounds__(256) vq_outputs(const float* __restrict__ in,
                                                  const float* __restrict__ emb,
                                                  const int* __restrict__ idxbuf,
                                                  float* __restrict__ quantized,
                                                  float* __restrict__ latents,
                                                  float* __restrict__ qlatents) {
    __shared__ float tile[64 * 65];   // +1 pad: conflict-free both directions
    __shared__ int   idx_s[64];
    const int t  = threadIdx.x;
    const int bh = blockIdx.x;        // 0..1023  (b = bh/64, h = bh%64)
    const int b  = bh >> 6;
    const int h  = bh & 63;

    if (t < 64) idx_s[t] = idxbuf[bh * 64 + t];

    // Phase 1: load input tile [c][w] (coalesced along w).
    #pragma unroll
    for (int i = 0; i < 16; ++i) {
        int lin = t + i * 256;
        int c = lin >> 6, w = lin & 63;
        tile[c * 65 + w] = in[b * CHW + c * HW + h * Wsz + w];
    }
    __syncthreads();

    // Phase 2: latents [b,h,w,c] (coalesced write, conflict-free LDS read).
    #pragma unroll
    for (int i = 0; i < 16; ++i) {
        int lin = t + i * 256;
        int w = lin >> 6, c = lin & 63;
        latents[(size_t)bh * 4096 + lin] = tile[c * 65 + w];
    }
    __syncthreads();

    // Phase 3: gather codebook rows; write qlatents, stage transpose.
    #pragma unroll
    for (int i = 0; i < 16; ++i) {
        int lin = t + i * 256;
        int w = lin >> 6, c = lin & 63;
        float q = emb[idx_s[w] * Csz + c];          // coalesced per w-row
        qlatents[(size_t)bh * 4096 + lin] = q;
        tile[w * 65 + c] = q;
    }
    __syncthreads();

    // Phase 4: quantized [b,c,h,w] (straight-through fwd value == gather).
    #pragma unroll
    for (int i = 0; i < 16; ++i) {
        int lin = t + i * 256;
        int c = lin >> 6, w = lin & 63;
        quantized[b * CHW + c * HW + h * Wsz + w] = tile[w * 65 + c];
    }
}

// ---------------------------------------------------------------------------
extern "C" void kernel_launch(void* const* d_in, const int* in_sizes, int n_in,
                              void* d_out, int out_size, void* d_ws, size_t ws_size,
                              hipStream_t stream) {
    const float* in  = (const float*)d_in[0];   // [B,C,H,W]
    const float* emb = (const float*)d_in[1];   // [K,C]

    float* norms = (float*)d_ws;                         // 4 KB
    int*   idxb  = (int*)((char*)d_ws + 4096);           // 256 KB

    float* out_q  = (float*)d_out;                       // quantized [B,C,H,W]
    float* out_l  = out_q + (size_t)Bsz * Csz * Hsz * Wsz; // latents [B,H,W,C]
    float* out_ql = out_l + (size_t)Bsz * Csz * Hsz * Wsz; // quantized_latents

    vq_norms<<<Ksz / 8, 256, 0, stream>>>(emb, norms);

    size_t lds2 = (size_t)(Ksz * Csz + Ksz) * sizeof(float);   // 260 KB < 320 KB
    vq_argmin<<<Nflat / (16 * 8), 256, lds2, stream>>>(in, emb, norms, idxb);

    vq_outputs<<<Bsz * Hsz, 256, 0, stream>>>(in, emb, idxb, out_q, out_l, out_ql);
}